// DotGATLayer_13872744366437
// MI455X (gfx1250) — compile-verified
//
#include <hip/hip_runtime.h>
#include <hip/hip_bf16.h>

typedef float v2f __attribute__((ext_vector_type(2)));
typedef float v8f __attribute__((ext_vector_type(8)));

// ---------------------------------------------------------------------------
// Init: zero h, denom; m_enc = 0 (acts as -inf under the monotonic encoding)
// ---------------------------------------------------------------------------
__global__ void gat_init_kernel(float* __restrict__ h,
                                unsigned* __restrict__ m_enc,
                                float* __restrict__ denom,
                                int n_nodes, int tot) {
  int i = blockIdx.x * blockDim.x + threadIdx.x;
  if (i < tot) h[i] = 0.0f;
  if (i < n_nodes) {
    m_enc[i] = 0u;
    denom[i] = 0.0f;
  }
}

// ---------------------------------------------------------------------------
// Projections: q = z*Wq + bq, k = z*Wk + bk, v = z*Wv + bv
// One wave (32 threads) per 16-node tile, fp32 WMMA 16x16x4.
// A layout (ISA 32-bit A 16x4): lane holds row M = lane%16;
//   elem0 = K = 2*(lane/16), elem1 = K = 2*(lane/16)+1 within each k-step of 4.
// B layout mirrors A with N = lane%16. C/D: VGPR g -> row g + 8*(lane/16).
// ---------------------------------------------------------------------------
__global__ __launch_bounds__(32) void gat_proj_kernel(
    const float* __restrict__ z,
    const float* __restrict__ Wq, const float* __restrict__ bq,
    const float* __restrict__ Wk, const float* __restrict__ bk,
    const float* __restrict__ Wv, const float* __restrict__ bv,
    float* __restrict__ q, float* __restrict__ k, float* __restrict__ v,
    int n_nodes) {
  const int lane = threadIdx.x;
  const int m0 = blockIdx.x * 16;
  if (m0 >= n_nodes) return;              // uniform per block: EXEC stays all-1s
  const int row = m0 + (lane & 15);
  const int hi = lane >> 4;               // 0: lanes 0-15, 1: lanes 16-31
  const int ncol = lane & 15;

  // Preload A fragments for all 16 k-steps (reused for all 3 matrices, 4 n-tiles)
  v2f afrag[16];
  const float* zrow = z + (size_t)row * 64;
#pragma unroll
  for (int j = 0; j < 16; ++j) {
    const int c = 4 * j + 2 * hi;
    afrag[j].x = zrow[c];
    afrag[j].y = zrow[c + 1];
  }

  const float* Ws[3] = {Wq, Wk, Wv};
  const float* Bs[3] = {bq, bk, bv};
  float* Os[3] = {q, k, v};

  for (int mtx = 0; mtx < 3; ++mtx) {
    const float* W = Ws[mtx];
    const float* bias = Bs[mtx];
    float* out = Os[mtx];
    for (int n0 = 0; n0 < 64; n0 += 16) {
      v8f acc = {};
#pragma unroll
      for (int j = 0; j < 16; ++j) {
        const int kk = 4 * j + 2 * hi;
        v2f bfrag;
        bfrag.x = W[(size_t)kk * 64 + n0 + ncol];
        bfrag.y = W[(size_t)(kk + 1) * 64 + n0 + ncol];
        acc = __builtin_amdgcn_wmma_f32_16x16x4_f32(
            /*neg_a=*/false, afrag[j], /*neg_b=*/false, bfrag,
            /*c_mod=*/(short)0, acc, /*reuse_a=*/false, /*reuse_b=*/false);
      }
      const float bval = bias[n0 + ncol];
#pragma unroll
      for (int g = 0; g < 8; ++g) {
        const int r = m0 + g + 8 * hi;
        out[(size_t)r * 64 + n0 + ncol] = acc[g] + bval;
      }
    }
  }
}

// ---------------------------------------------------------------------------
// Edge scores: e = <k[src], q[dst]> * tau ; atomic segment-max into m_enc[dst]
// 8 lanes per edge, coalesced float4 gathers (q/k are L2-resident: 51 MB).
// ---------------------------------------------------------------------------
__global__ void gat_edge_score_kernel(const float* __restrict__ q,
                                      const float* __restrict__ k,
                                      const int* __restrict__ src,
                                      const int* __restrict__ dst,
                                      float* __restrict__ e_out,
                                      unsigned* __restrict__ m_enc,
                                      int n_edges) {
  const int t = blockIdx.x * blockDim.x + threadIdx.x;
  const int edge = t >> 3;
  const int sub = t & 7;
  if (edge >= n_edges) return;
  const int s = src[edge];
  const int d = dst[edge];
  const float4* kp = (const float4*)(k + (size_t)s * 64) + sub * 2;
  const float4* qp = (const float4*)(q + (size_t)d * 64) + sub * 2;
  const float4 k0 = kp[0], k1 = kp[1];
  const float4 q0 = qp[0], q1 = qp[1];
  float p = k0.x * q0.x + k0.y * q0.y + k0.z * q0.z + k0.w * q0.w +
            k1.x * q1.x + k1.y * q1.y + k1.z * q1.z + k1.w * q1.w;
  p += __shfl_xor(p, 1);
  p += __shfl_xor(p, 2);
  p += __shfl_xor(p, 4);
  if (sub == 0) {
    const float e = p * 0.125f;  // tau = 1/sqrt(64)
    e_out[edge] = e;
    const unsigned u = __float_as_uint(e);
    const unsigned enc = (u & 0x80000000u) ? ~u : (u | 0x80000000u);
    atomicMax(m_enc + d, enc);
  }
}

// ---------------------------------------------------------------------------
// ex = exp(e - m[dst]); denom[dst] += ex ; stage ex in the alpha output slot
// ---------------------------------------------------------------------------
__global__ void gat_edge_exp_kernel(const float* __restrict__ e_in,
                                    const int* __restrict__ dst,
                                    const unsigned* __restrict__ m_enc,
                                    float* __restrict__ denom,
                                    float* __restrict__ ex_out,
                                    int n_edges) {
  const int i = blockIdx.x * blockDim.x + threadIdx.x;
  if (i >= n_edges) return;
  const int d = dst[i];
  const unsigned enc = m_enc[d];
  const unsigned u = (enc & 0x80000000u) ? (enc & 0x7FFFFFFFu) : ~enc;
  const float m = __uint_as_float(u);
  const float ex = __expf(e_in[i] - m);
  ex_out[i] = ex;
  unsafeAtomicAdd(denom + d, ex);
}

// ---------------------------------------------------------------------------
// alpha = ex / denom[dst]; h[dst] += alpha * v[src]   (8 lanes per edge)
// alpha buffer holds ex on entry; lane 0 overwrites it with normalized alpha.
// ---------------------------------------------------------------------------
__global__ void gat_edge_aggr_kernel(const float* __restrict__ v,
                                     const int* __restrict__ src,
                                     const int* __restrict__ dst,
                                     const float* __restrict__ denom,
                                     float* __restrict__ alpha,
                                     float* __restrict__ h,
                                     int n_edges) {
  const int t = blockIdx.x * blockDim.x + threadIdx.x;
  const int edge = t >> 3;
  const int sub = t & 7;
  if (edge >= n_edges) return;
  const int s = src[edge];
  const int d = dst[edge];
  const float a = alpha[edge] / denom[d];  // alpha[] currently holds ex
  const float4* vp = (const float4*)(v + (size_t)s * 64) + sub * 2;
  const float4 v0 = vp[0], v1 = vp[1];
  float* hp = h + (size_t)d * 64 + sub * 8;
  unsafeAtomicAdd(hp + 0, a * v0.x);
  unsafeAtomicAdd(hp + 1, a * v0.y);
  unsafeAtomicAdd(hp + 2, a * v0.z);
  unsafeAtomicAdd(hp + 3, a * v0.w);
  unsafeAtomicAdd(hp + 4, a * v1.x);
  unsafeAtomicAdd(hp + 5, a * v1.y);
  unsafeAtomicAdd(hp + 6, a * v1.z);
  unsafeAtomicAdd(hp + 7, a * v1.w);
  if (sub == 0) alpha[edge] = a;
}

// ---------------------------------------------------------------------------
extern "C" void kernel_launch(void* const* d_in, const int* in_sizes, int n_in,
                              void* d_out, int out_size, void* d_ws, size_t ws_size,
                              hipStream_t stream) {
  const float* z  = (const float*)d_in[0];
  const float* Wq = (const float*)d_in[1];
  const float* bq = (const float*)d_in[2];
  const float* Wk = (const float*)d_in[3];
  const float* bk = (const float*)d_in[4];
  const float* Wv = (const float*)d_in[5];
  const float* bv = (const float*)d_in[6];
  const int* src  = (const int*)d_in[7];
  const int* dst  = (const int*)d_in[8];

  const int n_nodes = in_sizes[0] / 64;
  const int n_edges = in_sizes[7];

  float* h     = (float*)d_out;                       // [n_nodes*64]
  float* alpha = h + (size_t)n_nodes * 64;            // [n_edges]

  float* ws = (float*)d_ws;
  float* q = ws;                                      // 3 x [n_nodes*64]
  float* k = q + (size_t)n_nodes * 64;
  float* v = k + (size_t)n_nodes * 64;
  float* e = v + (size_t)n_nodes * 64;                // [n_edges]
  unsigned* m_enc = (unsigned*)(e + n_edges);         // [n_nodes]
  float* denom = (float*)(m_enc + n_nodes);           // [n_nodes]

  const int tot = n_nodes * 64;
  gat_init_kernel<<<(tot + 255) / 256, 256, 0, stream>>>(h, m_enc, denom,
                                                         n_nodes, tot);
  gat_proj_kernel<<<(n_nodes + 15) / 16, 32, 0, stream>>>(
      z, Wq, bq, Wk, bk, Wv, bv, q, k, v, n_nodes);
  gat_edge_score_kernel<<<(n_edges * 8 + 255) / 256, 256, 0, stream>>>(
      q, k, src, dst, e, m_enc, n_edges);
  gat_edge_exp_kernel<<<(n_edges + 255) / 256, 256, 0, stream>>>(
      e, dst, m_enc, denom, alpha, n_edges);
  gat_edge_aggr_kernel<<<(n_edges * 8 + 255) / 256, 256, 0, stream>>>(
      v, src, dst, denom, alpha, h, n_edges);
}